// CotLayer_79912161509431
// MI455X (gfx1250) — compile-verified
//
#include <hip/hip_runtime.h>

// CotLayer (temporal) for MI455X / gfx1250, fp32 end-to-end.
// GEMM stages on V_WMMA_F32_16X16X4_F32 (exact fp32 accumulate).
// One wave32 per tile; M=pixels (2x16 tiles/wave to reuse B frags),
// N=out-channels, K=in-channels in chunks of 4.

#define HWSZ 4096
#define CDIM 128
#define NB2  32
#define NFR  8
#define EPSV 1e-5f

typedef float v2f __attribute__((ext_vector_type(2)));
typedef float v8f __attribute__((ext_vector_type(8)));

__device__ __forceinline__ v8f wmma4(v2f a, v2f b, v8f c) {
  // D = A(16x4) * B(4x16) + C(16x16), all fp32
  return __builtin_amdgcn_wmma_f32_16x16x4_f32(false, a, false, b,
                                               (short)0, c, false, false);
}

#define V8ZERO ((v8f){0.f,0.f,0.f,0.f,0.f,0.f,0.f,0.f})

// ---------------------------------------------------------------------------
// 1) key_embed: grouped temporal conv3d (k=3, groups=4) + BN + ReLU -> k2d
//    Each source frame is loaded once; contributions scattered into a 3-slot
//    rolling accumulator ring (output frames n = m-1, m, m+1).
//    grid (HW/16, 4 groups, 4 batch), block 32 (one wave)
// ---------------------------------------------------------------------------
__global__ __launch_bounds__(32)
void k_keyembed(const float* __restrict__ x, const float* __restrict__ wk,
                const float* __restrict__ gg, const float* __restrict__ gb,
                const float* __restrict__ gm, const float* __restrict__ gv,
                float* __restrict__ k2d) {
  const int lane = threadIdx.x;
  const int ll = lane & 15, hi = lane >> 4;
  const int base = blockIdx.x * 16;
  const int grp  = blockIdx.y;
  const int bb   = blockIdx.z;

  // Preload all B fragments: w_key[o][ci][kt]; lane->N, lane-half->K pair
  v2f bf[3][2][8];
#pragma unroll
  for (int kt = 0; kt < 3; ++kt)
#pragma unroll
    for (int nt = 0; nt < 2; ++nt) {
      const int o = grp * 32 + nt * 16 + ll;
#pragma unroll
      for (int kc = 0; kc < 8; ++kc) {
        const int ci = kc * 4 + 2 * hi;
        bf[kt][nt][kc].x = wk[(o * 32 + ci) * 3 + kt];
        bf[kt][nt][kc].y = wk[(o * 32 + ci + 1) * 3 + kt];
      }
    }
  float sc[2], of[2];
#pragma unroll
  for (int nt = 0; nt < 2; ++nt) {
    const int ch = grp * 32 + nt * 16 + ll;
    const float s = gg[ch] * rsqrtf(gv[ch] + EPSV);
    sc[nt] = s; of[nt] = gb[ch] - gm[ch] * s;
  }

  v8f acc[3][2];            // ring slot = n % 3
#pragma unroll
  for (int nt = 0; nt < 2; ++nt) acc[0][nt] = V8ZERO;   // n = 0

  auto finalize = [&](int n) {
    const int s = n % 3;
#pragma unroll
    for (int nt = 0; nt < 2; ++nt) {
      const int ch = grp * 32 + nt * 16 + ll;
      float* kout = k2d + ((size_t)((bb * NFR + n) * CDIM + ch)) * HWSZ
                        + base + 8 * hi;
#pragma unroll
      for (int v = 0; v < 8; ++v) {
        const float t = acc[s][nt][v] * sc[nt] + of[nt];
        kout[v] = t > 0.f ? t : 0.f;
      }
    }
  };

#pragma unroll
  for (int m = 0; m < NFR; ++m) {
    // Load A fragments for source frame m exactly once
    v2f a[8];
    const float* xf = x + ((size_t)((bb * NFR + m) * CDIM + grp * 32)) * HWSZ
                        + base + ll;
#pragma unroll
    for (int kc = 0; kc < 8; ++kc) {
      const int ci = kc * 4 + 2 * hi;
      a[kc].x = xf[(size_t)ci * HWSZ];
      a[kc].y = xf[(size_t)(ci + 1) * HWSZ];
    }
    if (m + 1 < NFR) {                 // open accumulator for n = m+1 (kt=0)
      const int s = (m + 1) % 3;
#pragma unroll
      for (int nt = 0; nt < 2; ++nt) acc[s][nt] = V8ZERO;
#pragma unroll
      for (int kc = 0; kc < 8; ++kc) {
        acc[s][0] = wmma4(a[kc], bf[0][0][kc], acc[s][0]);
        acc[s][1] = wmma4(a[kc], bf[0][1][kc], acc[s][1]);
      }
    }
    {                                  // n = m (kt=1)
      const int s = m % 3;
#pragma unroll
      for (int kc = 0; kc < 8; ++kc) {
        acc[s][0] = wmma4(a[kc], bf[1][0][kc], acc[s][0]);
        acc[s][1] = wmma4(a[kc], bf[1][1][kc], acc[s][1]);
      }
    }
    if (m >= 1) {                      // n = m-1 (kt=2), then complete
      const int s = (m - 1) % 3;
#pragma unroll
      for (int kc = 0; kc < 8; ++kc) {
        acc[s][0] = wmma4(a[kc], bf[2][0][kc], acc[s][0]);
        acc[s][1] = wmma4(a[kc], bf[2][1][kc], acc[s][1]);
      }
      finalize(m - 1);
    }
  }
  finalize(NFR - 1);
}

// ---------------------------------------------------------------------------
// 2) embed: e = ReLU(BN(W_e1 @ [x;k2d])) (256->64), wd = W_e2 @ e + b (64->96)
//    Two 16-pixel M-tiles per wave (B fragments reused 2x).
//    wd stored pixel-major (B2, HW, 96). grid (HW/32, 32), block 32.
// ---------------------------------------------------------------------------
__global__ __launch_bounds__(32)
void k_embed(const float* __restrict__ x, const float* __restrict__ k2d,
             const float* __restrict__ we1,
             const float* __restrict__ eg, const float* __restrict__ eb,
             const float* __restrict__ em, const float* __restrict__ ev,
             const float* __restrict__ we2, const float* __restrict__ be2,
             float* __restrict__ wdr) {
  __shared__ float lds_e[64 * 32];
  const int lane = threadIdx.x;
  const int ll = lane & 15, hi = lane >> 4;
  const int base = blockIdx.x * 32;
  const int b2 = blockIdx.y;

  const float* qk0 = x   + (size_t)b2 * CDIM * HWSZ + base + ll;
  const float* qk1 = k2d + (size_t)b2 * CDIM * HWSZ + base + ll;

  v8f acc[2][4];
#pragma unroll
  for (int t = 0; t < 2; ++t)
#pragma unroll
    for (int i = 0; i < 4; ++i) acc[t][i] = V8ZERO;

  for (int kc = 0; kc < 64; ++kc) {            // K = 256 channels
    const int c = kc * 4 + 2 * hi;
    const float* src = (kc < 32) ? (qk0 + (size_t)c * HWSZ)
                                 : (qk1 + (size_t)(c - 128) * HWSZ);
    v2f a0; a0.x = src[0];  a0.y = src[HWSZ];
    v2f a1; a1.x = src[16]; a1.y = src[HWSZ + 16];
#pragma unroll
    for (int nt = 0; nt < 4; ++nt) {
      const int o = nt * 16 + ll;
      v2f bv; bv.x = we1[o * 256 + c]; bv.y = we1[o * 256 + c + 1];
      acc[0][nt] = wmma4(a0, bv, acc[0][nt]);
      acc[1][nt] = wmma4(a1, bv, acc[1][nt]);
    }
  }
  // BN + ReLU, transpose to A-fragment layout via LDS
#pragma unroll
  for (int nt = 0; nt < 4; ++nt) {
    const int ch = nt * 16 + ll;
    const float s = eg[ch] * rsqrtf(ev[ch] + EPSV);
    const float o = eb[ch] - em[ch] * s;
#pragma unroll
    for (int t = 0; t < 2; ++t)
#pragma unroll
      for (int v = 0; v < 8; ++v) {
        const float tt = acc[t][nt][v] * s + o;
        lds_e[ch * 32 + t * 16 + v + 8 * hi] = tt > 0.f ? tt : 0.f;
      }
  }
  __syncthreads();

  v8f aw[2][6];
#pragma unroll
  for (int t = 0; t < 2; ++t)
#pragma unroll
    for (int i = 0; i < 6; ++i) aw[t][i] = V8ZERO;

  for (int kc = 0; kc < 16; ++kc) {            // K = 64
    const int k = kc * 4 + 2 * hi;
    v2f a0; a0.x = lds_e[k * 32 + ll];      a0.y = lds_e[(k + 1) * 32 + ll];
    v2f a1; a1.x = lds_e[k * 32 + 16 + ll]; a1.y = lds_e[(k + 1) * 32 + 16 + ll];
#pragma unroll
    for (int nt = 0; nt < 6; ++nt) {
      const int o = nt * 16 + ll;
      v2f bv; bv.x = we2[o * 64 + k]; bv.y = we2[o * 64 + k + 1];
      aw[0][nt] = wmma4(a0, bv, aw[0][nt]);
      aw[1][nt] = wmma4(a1, bv, aw[1][nt]);
    }
  }
#pragma unroll
  for (int nt = 0; nt < 6; ++nt) {
    const int ch = nt * 16 + ll;
    const float bias = be2[ch];
#pragma unroll
    for (int t = 0; t < 2; ++t)
#pragma unroll
      for (int v = 0; v < 8; ++v) {
        const int pix = base + t * 16 + v + 8 * hi;
        wdr[((size_t)b2 * HWSZ + pix) * 96 + ch] = aw[t][nt][v] + bias;
      }
  }
}

// ---------------------------------------------------------------------------
// 3) GroupNorm stats: mean / rsigma per (b2, group of 3 channels)
// ---------------------------------------------------------------------------
__global__ __launch_bounds__(256)
void k_gnstats(const float* __restrict__ wdr, float* __restrict__ st) {
  __shared__ float s0[256], s1[256];
  const int tid = threadIdx.x;
  const int b2 = blockIdx.x >> 5, c1 = blockIdx.x & 31;
  float sum = 0.f, sq = 0.f;
  for (int pix = tid; pix < HWSZ; pix += 256) {
    const float* p = wdr + ((size_t)b2 * HWSZ + pix) * 96 + c1 * 3;
#pragma unroll
    for (int k = 0; k < 3; ++k) { const float v = p[k]; sum += v; sq += v * v; }
  }
  s0[tid] = sum; s1[tid] = sq; __syncthreads();
  for (int off = 128; off > 0; off >>= 1) {
    if (tid < off) { s0[tid] += s0[tid + off]; s1[tid] += s1[tid + off]; }
    __syncthreads();
  }
  if (tid == 0) {
    const float inv = 1.f / (3.f * HWSZ);
    const float mu = s0[0] * inv;
    const float var = s1[0] * inv - mu * mu;
    st[blockIdx.x * 2]     = mu;
    st[blockIdx.x * 2 + 1] = rsqrtf(var + EPSV);
  }
}

// ---------------------------------------------------------------------------
// 4) value: v = BN(W_1x1 @ x)  (128->128 GEMM), 2 M-tiles/wave,
//    channel-major output. grid (HW/32, 32), block 32.
// ---------------------------------------------------------------------------
__global__ __launch_bounds__(32)
void k_value(const float* __restrict__ x, const float* __restrict__ w1,
             const float* __restrict__ vg, const float* __restrict__ vb,
             const float* __restrict__ vm, const float* __restrict__ vvv,
             float* __restrict__ vout) {
  __shared__ float lds_v[128 * 32];
  const int lane = threadIdx.x;
  const int ll = lane & 15, hi = lane >> 4;
  const int base = blockIdx.x * 32;
  const int b2 = blockIdx.y;
  const float* xp = x + (size_t)b2 * CDIM * HWSZ + base + ll;

  v8f acc[2][8];
#pragma unroll
  for (int t = 0; t < 2; ++t)
#pragma unroll
    for (int i = 0; i < 8; ++i) acc[t][i] = V8ZERO;

  for (int kc = 0; kc < 32; ++kc) {            // K = 128
    const int c = kc * 4 + 2 * hi;
    v2f a0; a0.x = xp[(size_t)c * HWSZ];      a0.y = xp[(size_t)(c + 1) * HWSZ];
    v2f a1; a1.x = xp[(size_t)c * HWSZ + 16]; a1.y = xp[(size_t)(c + 1) * HWSZ + 16];
#pragma unroll
    for (int nt = 0; nt < 8; ++nt) {
      const int o = nt * 16 + ll;
      v2f bv; bv.x = w1[o * 128 + c]; bv.y = w1[o * 128 + c + 1];
      acc[0][nt] = wmma4(a0, bv, acc[0][nt]);
      acc[1][nt] = wmma4(a1, bv, acc[1][nt]);
    }
  }
#pragma unroll
  for (int nt = 0; nt < 8; ++nt) {
    const int ch = nt * 16 + ll;
    const float s = vg[ch] * rsqrtf(vvv[ch] + EPSV);
    const float o = vb[ch] - vm[ch] * s;
#pragma unroll
    for (int t = 0; t < 2; ++t)
#pragma unroll
      for (int v = 0; v < 8; ++v)
        lds_v[ch * 32 + t * 16 + v + 8 * hi] = acc[t][nt][v] * s + o;
  }
  __syncthreads();
  // coalesced b128 stores: 4 channels per lane, 32 floats each
#pragma unroll
  for (int i = 0; i < 4; ++i) {
    const int ch = lane + 32 * i;
    const float4* srcv = (const float4*)(lds_v + ch * 32);
    float4* dst = (float4*)(vout + ((size_t)b2 * CDIM + ch) * HWSZ + base);
#pragma unroll
    for (int q = 0; q < 8; ++q) dst[q] = srcv[q];
  }
}

// ---------------------------------------------------------------------------
// 5) dynamic temporal aggregation + BN2 + SiLU -> agg
// ---------------------------------------------------------------------------
__global__ __launch_bounds__(256)
void k_aggregate(const float* __restrict__ wdr, const float* __restrict__ st,
                 const float* __restrict__ gng, const float* __restrict__ gnb,
                 const float* __restrict__ vbuf,
                 const float* __restrict__ g2, const float* __restrict__ b2c,
                 const float* __restrict__ m2, const float* __restrict__ v2c,
                 float* __restrict__ agg) {
  const int tid = threadIdx.x;
  const int b2 = blockIdx.y;
  const int pix = blockIdx.x * 256 + tid;
  const int bb = b2 >> 3, n = b2 & 7;
  const float* wrow = wdr + ((size_t)b2 * HWSZ + pix) * 96;

  for (int c1 = 0; c1 < 32; ++c1) {
    const float mu = st[(b2 * 32 + c1) * 2];
    const float rs = st[(b2 * 32 + c1) * 2 + 1];
    float wn[3];
#pragma unroll
    for (int k = 0; k < 3; ++k) {
      const int ch = c1 * 3 + k;
      wn[k] = (wrow[ch] - mu) * rs * gng[ch] + gnb[ch];
    }
#pragma unroll
    for (int c2 = 0; c2 < 4; ++c2) {
      const int c = c1 * 4 + c2;
      float s = 0.f;
#pragma unroll
      for (int k = 0; k < 3; ++k) {
        const int m = n + k - 1;
        if (m >= 0 && m < NFR)
          s += wn[k] * vbuf[((size_t)((bb * NFR + m) * CDIM + c)) * HWSZ + pix];
      }
      const float sc = g2[c] * rsqrtf(v2c[c] + EPSV);
      float a = s * sc + (b2c[c] - m2[c] * sc);
      a = a / (1.f + __expf(-a));              // SiLU
      agg[((size_t)b2 * CDIM + c) * HWSZ + pix] = a;
    }
  }
}

// ---------------------------------------------------------------------------
// 6) GAP over (n,h,w) of (agg + k2d) per (b, c)
// ---------------------------------------------------------------------------
__global__ __launch_bounds__(256)
void k_gap(const float* __restrict__ agg, const float* __restrict__ k2d,
           float* __restrict__ gap) {
  __shared__ float sm[256];
  const int tid = threadIdx.x;
  const int bb = blockIdx.x >> 7, c = blockIdx.x & 127;
  float sum = 0.f;
  for (int n = 0; n < NFR; ++n) {
    const size_t off = ((size_t)((bb * NFR + n) * CDIM + c)) * HWSZ;
    for (int pix = tid; pix < HWSZ; pix += 256)
      sum += agg[off + pix] + k2d[off + pix];
  }
  sm[tid] = sum; __syncthreads();
  for (int off = 128; off > 0; off >>= 1) {
    if (tid < off) sm[tid] += sm[tid + off];
    __syncthreads();
  }
  if (tid == 0) gap[blockIdx.x] = sm[0] / (float)(NFR * HWSZ);
}

// ---------------------------------------------------------------------------
// 7) SE MLP + radix-2 softmax (tiny)  block 512: tid = b*128 + c
// ---------------------------------------------------------------------------
__global__ __launch_bounds__(512)
void k_se(const float* __restrict__ gap,
          const float* __restrict__ ws1, const float* __restrict__ bs1,
          const float* __restrict__ sg, const float* __restrict__ sb,
          const float* __restrict__ smn, const float* __restrict__ sv,
          const float* __restrict__ ws2, const float* __restrict__ bs2,
          float* __restrict__ attn) {
  __shared__ float ash[512];
  const int tid = threadIdx.x;
  const int bb = tid >> 7, o = tid & 127;
  float h = bs1[o];
  for (int ci = 0; ci < 128; ++ci) h += gap[bb * 128 + ci] * ws1[o * 128 + ci];
  const float s = sg[o] * rsqrtf(sv[o] + EPSV);
  h = h * s + (sb[o] - smn[o] * s);
  h = h > 0.f ? h : 0.f;
  ash[tid] = h;
  __syncthreads();
  float l0 = bs2[2 * o], l1 = bs2[2 * o + 1];
  for (int j = 0; j < 128; ++j) {
    const float av = ash[bb * 128 + j];
    l0 += av * ws2[(2 * o) * 128 + j];
    l1 += av * ws2[(2 * o + 1) * 128 + j];
  }
  const float mx = l0 > l1 ? l0 : l1;
  const float e0 = __expf(l0 - mx), e1 = __expf(l1 - mx);
  const float inv = 1.f / (e0 + e1);
  attn[tid * 2]     = e0 * inv;
  attn[tid * 2 + 1] = e1 * inv;
}

// ---------------------------------------------------------------------------
// 8) out = agg*attn0 + k2d*attn1
// ---------------------------------------------------------------------------
__global__ __launch_bounds__(256)
void k_out(const float* __restrict__ agg, const float* __restrict__ k2d,
           const float* __restrict__ attn, float* __restrict__ out) {
  const int pix = blockIdx.x * 256 + threadIdx.x;
  const int c = blockIdx.y, b2 = blockIdx.z;
  const int bb = b2 >> 3;
  const float a0 = attn[(bb * 128 + c) * 2];
  const float a1 = attn[(bb * 128 + c) * 2 + 1];
  const size_t idx = ((size_t)b2 * CDIM + c) * HWSZ + pix;
  out[idx] = agg[idx] * a0 + k2d[idx] * a1;
}

// ---------------------------------------------------------------------------
extern "C" void kernel_launch(void* const* d_in, const int* in_sizes, int n_in,
                              void* d_out, int out_size, void* d_ws, size_t ws_size,
                              hipStream_t stream) {
  (void)in_sizes; (void)n_in; (void)out_size; (void)ws_size;
  const float* x     = (const float*)d_in[0];
  const float* w_key = (const float*)d_in[1];
  const float* bnk_g = (const float*)d_in[2];
  const float* bnk_b = (const float*)d_in[3];
  const float* bnk_m = (const float*)d_in[4];
  const float* bnk_v = (const float*)d_in[5];
  const float* w_e1  = (const float*)d_in[6];
  const float* bne_g = (const float*)d_in[7];
  const float* bne_b = (const float*)d_in[8];
  const float* bne_m = (const float*)d_in[9];
  const float* bne_v = (const float*)d_in[10];
  const float* w_e2  = (const float*)d_in[11];
  const float* b_e2  = (const float*)d_in[12];
  const float* gn_g  = (const float*)d_in[13];
  const float* gn_b  = (const float*)d_in[14];
  const float* w_1x1 = (const float*)d_in[15];
  const float* bn1_g = (const float*)d_in[16];
  const float* bn1_b = (const float*)d_in[17];
  const float* bn1_m = (const float*)d_in[18];
  const float* bn1_v = (const float*)d_in[19];
  const float* bn2_g = (const float*)d_in[20];
  const float* bn2_b = (const float*)d_in[21];
  const float* bn2_m = (const float*)d_in[22];
  const float* bn2_v = (const float*)d_in[23];
  const float* w_se1 = (const float*)d_in[24];
  const float* b_se1 = (const float*)d_in[25];
  const float* bnse_g = (const float*)d_in[26];
  const float* bnse_b = (const float*)d_in[27];
  const float* bnse_m = (const float*)d_in[28];
  const float* bnse_v = (const float*)d_in[29];
  const float* w_se2 = (const float*)d_in[30];
  const float* b_se2 = (const float*)d_in[31];
  float* out = (float*)d_out;

  float* ws   = (float*)d_ws;
  float* k2d  = ws;                                    // 32*128*4096
  float* wdr  = k2d  + (size_t)NB2 * CDIM * HWSZ;      // 32*4096*96
  float* vbuf = wdr  + (size_t)NB2 * HWSZ * 96;        // 32*128*4096
  float* agg  = vbuf + (size_t)NB2 * CDIM * HWSZ;      // 32*128*4096
  float* st   = agg  + (size_t)NB2 * CDIM * HWSZ;      // 1024*2
  float* gap  = st   + 2048;                           // 512
  float* attn = gap  + 512;                            // 1024

  k_keyembed<<<dim3(HWSZ / 16, 4, 4), 32, 0, stream>>>(
      x, w_key, bnk_g, bnk_b, bnk_m, bnk_v, k2d);
  k_embed<<<dim3(HWSZ / 32, NB2), 32, 0, stream>>>(
      x, k2d, w_e1, bne_g, bne_b, bne_m, bne_v, w_e2, b_e2, wdr);
  k_gnstats<<<1024, 256, 0, stream>>>(wdr, st);
  k_value<<<dim3(HWSZ / 32, NB2), 32, 0, stream>>>(
      x, w_1x1, bn1_g, bn1_b, bn1_m, bn1_v, vbuf);
  k_aggregate<<<dim3(HWSZ / 256, NB2), 256, 0, stream>>>(
      wdr, st, gn_g, gn_b, vbuf, bn2_g, bn2_b, bn2_m, bn2_v, agg);
  k_gap<<<512, 256, 0, stream>>>(agg, k2d, gap);
  k_se<<<1, 512, 0, stream>>>(
      gap, w_se1, b_se1, bnse_g, bnse_b, bnse_m, bnse_v, w_se2, b_se2, attn);
  k_out<<<dim3(HWSZ / 256, CDIM, NB2), 256, 0, stream>>>(agg, k2d, attn, out);
}